// TrajLoss_54168127537192
// MI455X (gfx1250) — compile-verified
//
#include <hip/hip_runtime.h>
#include <math.h>

typedef float v2f __attribute__((ext_vector_type(2)));
typedef float v4f __attribute__((ext_vector_type(4)));
typedef float v8f __attribute__((ext_vector_type(8)));

#define NPTS 4096
#define DIM  32
#define TILES (NPTS / 16)   // 256 row tiles of 16
#define NW 8                // waves per block (256 threads, wave32)
#define NIT (TILES / NW)    // 32 tiles per wave

__device__ __forceinline__ void lse_merge(float& m, float& s, float m2, float s2) {
    float nm = fmaxf(m, m2);
    s = s * __expf(m - nm) + s2 * __expf(m2 - nm);
    m = nm;
}

__device__ __forceinline__ v2f load2(const float* p) {
    return *reinterpret_cast<const v2f*>(p);
}
__device__ __forceinline__ v4f load4(const float* p) {
    return *reinterpret_cast<const v4f*>(p);
}

// Fused "flash-LSE" pass over the implicit cost matrix.
//   C_or = 0.5*||out_o||^2 + 0.5*||red_r||^2 - out_o . red_r
//   ured_r := vred_r - 0.5*||red_r||^2   (precomputed by the producing pass)
//   out_o  = signA*eps*( logConst + LSE_r[(ured_r + out_o.red_r)/eps] - 0.5*||out_o||^2/eps )
//   uout_o = out_o - 0.5*||out_o||^2     (companion for the consuming pass)
// eps = useSched ? max(epsTarget, 2*maxC*powfac) : epsTarget
__global__ __launch_bounds__(256)
void lse_pass(float* __restrict__ out,
              float* __restrict__ uout,
              const float* __restrict__ outPts,
              const float* __restrict__ redPts,
              const float* __restrict__ out2,
              const float* __restrict__ ured,
              const float* __restrict__ maxCptr,
              float powfac, float epsTarget, int useSched,
              float signA, float logConst)
{
    float eps = epsTarget;
    if (useSched) eps = fmaxf(epsTarget, 2.0f * maxCptr[0] * powfac);
    const float inv_eps = 1.0f / eps;

    const int tid  = threadIdx.x;
    const int lane = tid & 31;
    const int wv   = tid >> 5;
    const int lo   = lane & 15;
    const int hi   = lane >> 4;            // 0 or 1
    const int colBase = blockIdx.x * 16;

    // B tile (16 output points x 32 dims) in registers for the whole kernel.
    // f32 16x16x4 WMMA operand layout: lane l<16 -> (K=k0,k0+1), lane l>=16 -> (K=k0+2,k0+3)
    v2f btile[8];
    const float* brow = outPts + (size_t)(colBase + lo) * DIM + 2 * hi;
#pragma unroll
    for (int s = 0; s < 8; ++s) btile[s] = load2(brow + 4 * s);

    // ---- software-pipelined row-tile loop (A double-buffered, u single-buffered).
    // NOT unrolled: every body consumes A loaded one full iteration earlier (across
    // the back-edge), which is the schedule the compiler pipelines with partial waits.
    v2f aC[8];
    {
        const float* arow = redPts + (size_t)(wv * 16 + lo) * DIM + 2 * hi;
#pragma unroll
        for (int s = 0; s < 8; ++s) aC[s] = load2(arow + 4 * s);
    }

    float mrun = -INFINITY, srun = 0.0f;

#pragma unroll 1
    for (int it = 0; it < NIT; ++it) {
        const int tile = it * NW + wv;
        const int nt   = (it + 1 < NIT) ? (tile + NW) : tile;   // clamp (redundant reload on last)

        // (1) u for CURRENT tile, issued first so a partial wait can release it
        //     before the next-tile A loads complete (loads retire in order).
        const int rbase = tile * 16 + (hi << 3);
        v4f u0 = load4(ured + rbase);
        v4f u1 = load4(ured + rbase + 4);

        // (2) prefetch next tile's A fragments
        v2f aN[8];
        {
            const float* narow = redPts + (size_t)(nt * 16 + lo) * DIM + 2 * hi;
#pragma unroll
            for (int s = 0; s < 8; ++s) aN[s] = load2(narow + 4 * s);
        }

        // (3) 16x16 Gram tile: 8 chained f32 WMMAs over K=32 (data loaded last iteration)
        v8f acc = {};
#pragma unroll
        for (int s = 0; s < 8; ++s)
            acc = __builtin_amdgcn_wmma_f32_16x16x4_f32(false, aC[s], false, btile[s],
                                                        (short)0, acc, false, false);

        // (4) online LSE epilogue.
        // D layout: lane l holds column colBase+(l&15); acc[r] is row tile*16 + r + 8*hi
        float tv[8];
        float tmax = -INFINITY;
#pragma unroll
        for (int r = 0; r < 8; ++r) {
            const float u = (r < 4) ? u0[r] : u1[r - 4];
            tv[r] = (u + acc[r]) * inv_eps;
            tmax = fmaxf(tmax, tv[r]);
        }
        float nm = fmaxf(mrun, tmax);
        float add = 0.0f;
#pragma unroll
        for (int r = 0; r < 8; ++r) add += __expf(tv[r] - nm);
        srun = srun * __expf(mrun - nm) + add;
        mrun = nm;

        // (5) rotate A double buffer (16 dual-issuable v_movs per tile)
#pragma unroll
        for (int s = 0; s < 8; ++s) aC[s] = aN[s];
    }

    // merge the two half-wave row groups (same column, different rows)
    lse_merge(mrun, srun, __shfl_xor(mrun, 16, 32), __shfl_xor(srun, 16, 32));

    __shared__ float smM[NW][16];
    __shared__ float smS[NW][16];
    if (lane < 16) { smM[wv][lane] = mrun; smS[wv][lane] = srun; }
    __syncthreads();
    if (tid < 16) {
        float M = smM[0][tid], S = smS[0][tid];
#pragma unroll
        for (int w = 1; w < NW; ++w) lse_merge(M, S, smM[w][tid], smS[w][tid]);
        const int col = colBase + tid;
        const float h2 = 0.5f * out2[col];
        const float L = M + __logf(S) - h2 * inv_eps;   // add per-column constant
        const float o = signA * eps * (logConst + L);
        out[col]  = o;
        uout[col] = o - h2;                             // companion for the next pass
    }
}

// max over C (for eps0 = 2*max(C)); runs only 4x total, un-pipelined
__global__ __launch_bounds__(256)
void cost_max(const float* __restrict__ outPts, const float* __restrict__ redPts,
              const float* __restrict__ out2, const float* __restrict__ red2,
              float* __restrict__ maxCptr)
{
    const int tid  = threadIdx.x;
    const int lane = tid & 31;
    const int wv   = tid >> 5;
    const int lo   = lane & 15;
    const int hi   = lane >> 4;
    const int colBase = blockIdx.x * 16;

    v2f btile[8];
    const float* brow = outPts + (size_t)(colBase + lo) * DIM + 2 * hi;
#pragma unroll
    for (int s = 0; s < 8; ++s) btile[s] = load2(brow + 4 * s);

    float lm = -INFINITY;
    for (int tile = wv; tile < TILES; tile += NW) {
        const int rowBase = tile * 16;
        const float* arow = redPts + (size_t)(rowBase + lo) * DIM + 2 * hi;
        v8f acc = {};
#pragma unroll
        for (int s = 0; s < 8; ++s) {
            v2f a = load2(arow + 4 * s);
            acc = __builtin_amdgcn_wmma_f32_16x16x4_f32(false, a, false, btile[s],
                                                        (short)0, acc, false, false);
        }
        const int rbase = rowBase + (hi << 3);
#pragma unroll
        for (int r = 0; r < 8; ++r)
            lm = fmaxf(lm, 0.5f * red2[rbase + r] - acc[r]);
    }
    float v = lm + 0.5f * out2[colBase + lo];
#pragma unroll
    for (int off = 16; off > 0; off >>= 1)
        v = fmaxf(v, __shfl_xor(v, off, 32));
    if (lane == 0)
        atomicMax(reinterpret_cast<unsigned int*>(maxCptr),
                  __float_as_uint(fmaxf(v, 0.0f)));
}

// row squared norms for x and obs; also onu = -0.5*||obs||^2 (the "u" vector for
// the fit pass-1 where vred == 0).
__global__ void norms_kernel(const float* __restrict__ x, const float* __restrict__ obs,
                             float* __restrict__ xn, float* __restrict__ on,
                             float* __restrict__ onu)
{
    int i = blockIdx.x * 256 + threadIdx.x;
    if (i >= 2 * 5 * NPTS) return;
    const bool isx = (i < 5 * NPTS);
    const int r = isx ? i : i - 5 * NPTS;
    const float* p = (isx ? x : obs) + (size_t)r * DIM;
    float s = 0.0f;
#pragma unroll
    for (int k = 0; k < DIM; ++k) { float v = p[k]; s += v * v; }
    if (isx) { xn[r] = s; } else { on[r] = s; onu[r] = -0.5f * s; }
}

__global__ void init_globals(float* regA, float* fitA)
{
    if (blockIdx.x == 0 && threadIdx.x == 0) { regA[0] = 0.0f; fitA[0] = 0.0f; }
}

// f = g = 0; fu = -0.5*||X||^2; gu = -0.5*||Y||^2; maxC = 0
__global__ void init_pair(float* f, float* fu, float* g, float* gu,
                          const float* __restrict__ X2, const float* __restrict__ Y2,
                          float* maxC)
{
    int i = blockIdx.x * 256 + threadIdx.x;
    if (i < NPTS) {
        f[i] = 0.0f; g[i] = 0.0f;
        fu[i] = -0.5f * X2[i];
        gu[i] = -0.5f * Y2[i];
    }
    if (i == 0) maxC[0] = 0.0f;
}

__global__ void reduce_mean_add(float* __restrict__ acc, const float* __restrict__ v,
                                int n, float scale)
{
    __shared__ float sm[256];
    float s = 0.0f;
    for (int i = threadIdx.x; i < n; i += 256) s += v[i];
    sm[threadIdx.x] = s;
    __syncthreads();
    for (int off = 128; off > 0; off >>= 1) {
        if (threadIdx.x < off) sm[threadIdx.x] += sm[threadIdx.x + off];
        __syncthreads();
    }
    if (threadIdx.x == 0) acc[0] += sm[0] * scale;   // single block -> deterministic
}

__global__ void finalize_kernel(float* __restrict__ out,
                                const float* __restrict__ regA,
                                const float* __restrict__ fitA)
{
    if (threadIdx.x == 0 && blockIdx.x == 0) {
        // LAM_REG * (EPS * sum_sinkhorn) + LAM_FIT * (N/10) * sum_nll
        out[0] = 0.05f * regA[0] + ((float)NPTS / 10.0f) * fitA[0];
    }
}

extern "C" void kernel_launch(void* const* d_in, const int* in_sizes, int n_in,
                              void* d_out, int out_size, void* d_ws, size_t ws_size,
                              hipStream_t stream)
{
    const float* x   = (const float*)d_in[0];   // (5, 4096, 32) f32
    const float* obs = (const float*)d_in[1];   // (5, 4096, 32) f32
    float* out = (float*)d_out;
    float* ws  = (float*)d_ws;

    float* f     = ws;                  // 4096
    float* fu    = ws + 4096;           // 4096  (f - 0.5*||X||^2)
    float* g     = ws + 8192;           // 4096
    float* gu    = ws + 12288;          // 4096  (g - 0.5*||Y||^2)
    float* bufA  = ws + 16384;          // 4096
    float* bufAu = ws + 20480;          // 4096
    float* bufB  = ws + 24576;          // 4096
    float* bufBu = ws + 28672;          // 4096
    float* xn    = ws + 32768;          // 5*4096 row sq-norms of x
    float* on    = ws + 53248;          // 5*4096 row sq-norms of obs
    float* onu   = ws + 73728;          // 5*4096: -0.5*on
    float* maxC  = ws + 94208;          // 1 (float bits via uint atomicMax)
    float* regA  = ws + 94209;          // 1
    float* fitA  = ws + 94210;          // 1

    const float eps_t = 0.05f;                  // blur^2 = EPS
    const float loga  = -logf((float)NPTS);     // == logb (n == m)

    init_globals<<<1, 64, 0, stream>>>(regA, fitA);
    norms_kernel<<<160, 256, 0, stream>>>(x, obs, xn, on, onu);

    // ---- PathsLoss: 4 consecutive-timepoint Sinkhorn costs ----
    for (int t = 0; t < 4; ++t) {
        const float* X  = x + (size_t)t * NPTS * DIM;
        const float* Y  = x + (size_t)(t + 1) * NPTS * DIM;
        const float* X2 = xn + (size_t)t * NPTS;
        const float* Y2 = xn + (size_t)(t + 1) * NPTS;

        init_pair<<<16, 256, 0, stream>>>(f, fu, g, gu, X2, Y2, maxC);
        cost_max<<<TILES, 256, 0, stream>>>(Y, X, Y2, X2, maxC);

        for (int k = 0; k < 80; ++k) {
            float pf = (float)pow(0.95, 2.0 * (double)k);
            // g = -eps*(loga + LSE_i[(f_i - C_ij)/eps])   (reduce over X rows, u = fu)
            lse_pass<<<TILES, 256, 0, stream>>>(g, gu, Y, X, Y2, fu, maxC,
                                                pf, eps_t, 1, -1.0f, loga);
            // f = -eps*(logb + LSE_j[(g_j - C_ij)/eps])   (reduce over Y rows, u = gu)
            lse_pass<<<TILES, 256, 0, stream>>>(f, fu, X, Y, X2, gu, maxC,
                                                pf, eps_t, 1, -1.0f, loga);
        }
        // differentiable extrapolation at target eps (forward value only)
        lse_pass<<<TILES, 256, 0, stream>>>(bufA, bufAu, Y, X, Y2, fu, maxC,
                                            0.0f, eps_t, 0, -1.0f, loga);  // g_f
        lse_pass<<<TILES, 256, 0, stream>>>(bufB, bufBu, X, Y, X2, gu, maxC,
                                            0.0f, eps_t, 0, -1.0f, loga);  // f_f
        reduce_mean_add<<<1, 256, 0, stream>>>(regA, bufA, NPTS, 1.0f / NPTS);
        reduce_mean_add<<<1, 256, 0, stream>>>(regA, bufB, NPTS, 1.0f / NPTS);
    }

    // ---- FitLoss per timepoint ----
    for (int t = 0; t < 5; ++t) {
        const float* X  = x   + (size_t)t * NPTS * DIM;
        const float* O  = obs + (size_t)t * NPTS * DIM;
        const float* X2 = xn  + (size_t)t * NPTS;
        const float* O2 = on  + (size_t)t * NPTS;
        const float* OU = onu + (size_t)t * NPTS;
        // pass 1: bufA_i = -LSE_j(logits_ij); logits = -0.5*sqdist; u = -0.5*||O||^2
        lse_pass<<<TILES, 256, 0, stream>>>(bufA, bufAu, X, O, X2, OU, maxC,
                                            0.0f, 1.0f, 0, -1.0f, 0.0f);
        // pass 2: bufB_j = LSE_i(logits_ij - rowLSE_i); u = bufAu = -rowLSE - 0.5*||X||^2
        lse_pass<<<TILES, 256, 0, stream>>>(bufB, bufBu, O, X, O2, bufAu, maxC,
                                            0.0f, 1.0f, 0, 1.0f, 0.0f);
        // nll_t = -mean_j(bufB_j)   (the -log(m/n) term is 0 since n == m)
        reduce_mean_add<<<1, 256, 0, stream>>>(fitA, bufB, NPTS, -1.0f / NPTS);
    }

    finalize_kernel<<<1, 1, 0, stream>>>(out, regA, fitA);
}